// YoloLayer_50895362458123
// MI455X (gfx1250) — compile-verified
//
#include <hip/hip_runtime.h>
#include <math.h>

#define NB     64
#define NA     3
#define NCLS   13
#define NH     76
#define NW     76
#define NLL    12
#define MAXT   50
#define NPIX   (NH*NW)          // 5776
#define NANCH  (NA*NPIX)        // 17328
#define CAD    (NB*NANCH)       // 1108992
#define CHT    (NLL+NCLS)       // 25 channels
#define NETWH  608.0f
#define BETA_F 0.028f
#define BLK    256
#define NBLK   (CAD/BLK)        // 4332 (exact)
#define RECS   16               // record stride (floats)
#define TPS    8                // tprep stride (floats), 32B rows for b128 async copy
#define NACC   10
#define PSTR   16               // partials stride (doubles)

static_assert(CAD % BLK == 0, "grid must tile exactly");

__device__ __forceinline__ float sigm(float x) { return 1.0f / (1.0f + expf(-x)); }

__device__ __forceinline__ float bcef(float p, float t) {
    float pc = fminf(fmaxf(p, 1e-12f), 1.0f);
    float qc = fminf(fmaxf(1.0f - p, 1e-12f), 1.0f);
    return -(t * logf(pc) + (1.0f - t) * logf(qc));
}

// ---------------------------------------------------------------- k_init
__global__ void k_init(int* __restrict__ tslot, int* __restrict__ clsmask) {
    int i = blockIdx.x * blockDim.x + threadIdx.x;
    if (i < CAD) { tslot[i] = -1; clsmask[i] = 0; }
}

// ---------------------------------------------------------------- k_targets
// One thread per batch; sequential over t => last-writer-wins scatter order.
__global__ void k_targets(const float* __restrict__ out, const float* __restrict__ tgt,
                          const float* __restrict__ anch, int* __restrict__ tslot,
                          int* __restrict__ clsmask, float* __restrict__ rec,
                          float* __restrict__ tprep, int* __restrict__ nvalid) {
    int b = blockIdx.x * blockDim.x + threadIdx.x;
    if (b >= NB) return;
    float aw[3] = {anch[0], anch[2], anch[4]};
    float ah[3] = {anch[1], anch[3], anch[5]};
    const float* tb = tgt + (size_t)b * MAXT * NLL;
    int nv = 0;
    for (int t = 0; t < MAXT; ++t) {
        const float* r = tb + t * NLL;
        float x = r[1];
        if (!(x > 0.0f)) break;               // valid = cumprod(x > 0) prefix
        float y  = r[2];
        float gx = x * NW, gy = y * NH;
        float gw = r[10] * NETWH, gh = r[11] * NETWH;
        float garea = gw * gh;
        int best = 0; float bestr = -1e30f;
        for (int a = 0; a < 3; ++a) {         // argmax: first max wins (strict >)
            float inter = fminf(gw, aw[a]) * fminf(gh, ah[a]);
            float ratio = inter / (garea + aw[a] * ah[a] - inter);
            if (ratio > bestr) { bestr = ratio; best = a; }
        }
        int gi = (int)floorf(gx); gi = gi < 0 ? 0 : (gi > NW - 1 ? NW - 1 : gi);
        int gj = (int)floorf(gy); gj = gj < 0 ? 0 : (gj > NH - 1 ? NH - 1 : gj);
        int flat = b * NANCH + best * NPIX + gj * NW + gi;
        size_t base = (size_t)(b * NA + best) * CHT * NPIX + (size_t)(gj * NW + gi);
        __builtin_prefetch(&out[base + (size_t)9 * NPIX], 0, 0);   // global_prefetch_b8
        float px = sigm(out[base]) + (float)gi;
        float py = sigm(out[base + (size_t)NPIX]) + (float)gj;
        float pw = expf(out[base + (size_t)9  * NPIX]) * aw[best];
        float ph = expf(out[base + (size_t)10 * NPIX]) * ah[best];
        // iou_t = iou(gt, pred_sel)
        float cw  = fminf(gx + 0.5f * gw, px + 0.5f * pw) - fmaxf(gx - 0.5f * gw, px - 0.5f * pw);
        float chh = fminf(gy + 0.5f * gh, py + 0.5f * ph) - fmaxf(gy - 0.5f * gh, py - 0.5f * ph);
        float inter = (cw > 0.0f && chh > 0.0f) ? cw * chh : 0.0f;
        float iou_t = inter / (garea + pw * ph - inter);
        int ri = b * MAXT + t;
        float* rr = rec + (size_t)ri * RECS;
        rr[0] = gx - (float)gi; rr[1] = gy - (float)gj;
        for (int k = 0; k < 7; ++k) rr[2 + k] = r[3 + k];          // tbox cols 3..9
        rr[9]  = logf(gw / aw[best]);
        rr[10] = logf(gh / ah[best]);
        rr[11] = iou_t;
        float* tp = tprep + (size_t)ri * TPS;
        tp[0] = gx - 0.5f * gw; tp[1] = gx + 0.5f * gw;
        tp[2] = gy - 0.5f * gh; tp[3] = gy + 0.5f * gh;
        tp[4] = garea; tp[5] = 0.0f; tp[6] = 0.0f; tp[7] = 0.0f;
        tslot[flat] = ri;                                          // last-wins
        clsmask[flat] |= (1 << (int)r[0]);                         // multi-set one-hot
        ++nv;
    }
    nvalid[b] = nv;
}

// ---------------------------------------------------------------- k_main
__global__ void __launch_bounds__(BLK) k_main(
        const float* __restrict__ out, const float* __restrict__ anch,
        const int* __restrict__ tslot, const int* __restrict__ clsmask,
        const float* __restrict__ rec, const float* __restrict__ tprep,
        const int* __restrict__ nvalid, double* __restrict__ partials) {
    __shared__ __align__(16) float sT[2 * MAXT * TPS];   // 2 batches x 50 targets x 32B
    __shared__ int    sNV[2];
    __shared__ double sW[BLK / 32][NACC];
    const int tid = threadIdx.x;
    const int i   = blockIdx.x * BLK + tid;
    const int b   = i / NANCH;
    const int rem = i - b * NANCH;
    const int a   = rem / NPIX;
    const int pix = rem - a * NPIX;
    const int bFirst = (blockIdx.x * BLK) / NANCH;
    const int bLast  = (blockIdx.x * BLK + (BLK - 1)) / NANCH;

    if (tid < 2) sNV[tid] = nvalid[tid == 0 ? bFirst : bLast];
    // --- async DMA of the two target tables into LDS (16B per lane) ---
    // 2 sections x MAXT rows x 32B = 2 x 100 x 16B chunks = 200 lanes.
    if (tid < 2 * MAXT * 2) {
        int s  = tid / (MAXT * 2);            // which staged batch section
        int j  = tid - s * (MAXT * 2);        // 16B chunk within section
        int bb = (s == 0) ? bFirst : bLast;
        const float* gsrc = tprep + (size_t)bb * MAXT * TPS + (size_t)j * 4;
        unsigned int lds_off = (unsigned int)(size_t)(void*)&sT[s * MAXT * TPS + j * 4];
        unsigned long long gaddr = (unsigned long long)(size_t)gsrc;
        asm volatile("global_load_async_to_lds_b128 %0, %1, off"
                     :: "v"(lds_off), "v"(gaddr) : "memory");
    }
    asm volatile("s_wait_asynccnt 0" ::: "memory");
    __syncthreads();

    const int sel = (b == bFirst) ? 0 : 1;
    const int nv  = sNV[sel];
    const float* st = sT + sel * MAXT * TPS;

    const size_t base = (size_t)(b * NA + a) * CHT * NPIX + (size_t)pix;
    const int slot = tslot[i];
    if (slot >= 0) {           // overlap obj-cell cacheline fetch with IoU loop
        __builtin_prefetch(&out[base + (size_t)2  * NPIX], 0, 0);
        __builtin_prefetch(&out[base + (size_t)5  * NPIX], 0, 0);
        __builtin_prefetch(&out[base + (size_t)12 * NPIX], 0, 0);
        __builtin_prefetch(&out[base + (size_t)18 * NPIX], 0, 0);
        __builtin_prefetch(&out[base + (size_t)24 * NPIX], 0, 0);
    }
    const float aw = anch[2 * a], ah = anch[2 * a + 1];
    const float v0  = out[base];
    const float v1  = out[base + (size_t)NPIX];
    const float v9  = out[base + (size_t)9  * NPIX];
    const float v10 = out[base + (size_t)10 * NPIX];
    const float vcf = out[base + (size_t)11 * NPIX];
    const int gyc = pix / NW, gxc = pix - gyc * NW;
    const float s0 = sigm(v0), s1 = sigm(v1);
    const float px = s0 + (float)gxc, py = s1 + (float)gyc;
    const float pw = expf(v9) * aw,  ph = expf(v10) * ah;
    const float pxl = px - 0.5f * pw, pxh = px + 0.5f * pw;
    const float pyl = py - 0.5f * ph, pyh = py + 0.5f * ph;
    const float pa  = pw * ph;
    bool hit = false;                 // iou>0.5  <=>  3*inter > pa+ta   (no div)
    for (int t = 0; t < nv; ++t) {
        const float4 tb4 = *reinterpret_cast<const float4*>(&st[t * TPS]); // ds_load_b128
        const float  ta  = st[t * TPS + 4];                                // ds_load_b32
        float cw  = fminf(pxh, tb4.y) - fmaxf(pxl, tb4.x);
        float chh = fminf(pyh, tb4.w) - fmaxf(pyl, tb4.z);
        float inter = (cw > 0.0f && chh > 0.0f) ? cw * chh : 0.0f;
        hit = hit || (3.0f * inter > pa + ta);
    }
    const float conf = sigm(vcf);
    float accC = 0.f, accConf = 0.f, accRot = 0.f, accCls = 0.f;
    float sq2 = 0.f, sq3 = 0.f, sq4 = 0.f, ab2 = 0.f, ab3 = 0.f, ab4 = 0.f;
    if (slot >= 0) {
        const float* rr = rec + (size_t)slot * RECS;
        accConf = bcef(conf, rr[11]);                              // conf_mask = 1
        accC = bcef(s0, rr[0]) + bcef(s1, rr[1]);
        float d9 = v9 - rr[9], d10 = v10 - rr[10];
        accC += d9 * d9 + d10 * d10;
        float v2 = out[base + (size_t)2 * NPIX];
        float v3 = out[base + (size_t)3 * NPIX];
        float v4 = out[base + (size_t)4 * NPIX];
        float d2 = v2 - rr[2], d3 = v3 - rr[3], d4 = v4 - rr[4];
        sq2 = d2*d2; sq3 = d3*d3; sq4 = d4*d4;
        ab2 = fabsf(d2); ab3 = fabsf(d3); ab4 = fabsf(d4);
        float v5 = out[base + (size_t)5 * NPIX];
        float v6 = out[base + (size_t)6 * NPIX];
        float v7 = out[base + (size_t)7 * NPIX];
        float v8 = out[base + (size_t)8 * NPIX];
        float nrm = fmaxf(sqrtf(v5*v5 + v6*v6 + v7*v7 + v8*v8), 1e-12f);
        accRot = fabsf(v5/nrm - rr[5]) + fabsf(v6/nrm - rr[6])
               + fabsf(v7/nrm - rr[7]) + fabsf(v8/nrm - rr[8]);
        int cm = clsmask[i];
        for (int c = 0; c < NCLS; ++c) {
            float x  = out[base + (size_t)(NLL + c) * NPIX];
            float tt = (float)((cm >> c) & 1);
            accCls += fmaxf(x, 0.0f) - x * tt + log1pf(expf(-fabsf(x)));
        }
    } else {
        accConf = hit ? 0.0f : bcef(conf, 0.0f);                   // noobj path
    }
    // deterministic reduction: wave32 butterfly -> 8 leaders -> 1 barrier -> 10 lanes
    double vals[NACC] = {(double)accC, (double)sq2, (double)sq3, (double)sq4,
                         (double)ab2, (double)ab3, (double)ab4,
                         (double)accRot, (double)accConf, (double)accCls};
    const int lane = tid & 31, wv = tid >> 5;
    #pragma unroll
    for (int k = 0; k < NACC; ++k) {
        double v = vals[k];
        #pragma unroll
        for (int off = 16; off > 0; off >>= 1) v += __shfl_xor(v, off, 32);
        if (lane == 0) sW[wv][k] = v;
    }
    __syncthreads();
    if (tid < NACC) {
        double s = 0.0;
        #pragma unroll
        for (int w8 = 0; w8 < BLK / 32; ++w8) s += sW[w8][tid];    // fixed order
        partials[(size_t)blockIdx.x * PSTR + tid] = s;
    }
}

// ---------------------------------------------------------------- k_final
__global__ void k_final(const double* __restrict__ partials, float* __restrict__ outv) {
    __shared__ double sred[BLK];
    __shared__ double tot[NACC];
    const int tid = threadIdx.x;
    for (int k = 0; k < NACC; ++k) {
        double s = 0.0;
        for (int j = tid; j < NBLK; j += BLK) s += partials[(size_t)j * PSTR + k];
        sred[tid] = s;
        __syncthreads();
        for (int st = BLK / 2; st > 0; st >>= 1) {
            if (tid < st) sred[tid] += sred[tid + st];
            __syncthreads();
        }
        if (tid == 0) tot[k] = sred[0];
        __syncthreads();
    }
    if (tid == 0) {
        const double invB = 1.0 / (double)NB;
        const double beta = (double)BETA_F;
        double loss = tot[0] * invB;                       // loss_coord (bce xy + sq wh)
        for (int c = 0; c < 3; ++c) {                      // loss_trans, global L2 branch
            double sq = tot[1 + c], ab = tot[4 + c];
            double dis = sqrt(sq);
            loss += ((dis <= beta) ? (0.5 * sq / beta) : (ab - beta * 0.5)) * invB;
        }
        loss += tot[7] * invB;                             // loss_rot
        loss += tot[8] * invB;                             // loss_conf
        loss += tot[9] * invB;                             // loss_cls
        outv[0] = (float)loss;
    }
}

// ---------------------------------------------------------------- launch
extern "C" void kernel_launch(void* const* d_in, const int* in_sizes, int n_in,
                              void* d_out, int out_size, void* d_ws, size_t ws_size,
                              hipStream_t stream) {
    (void)in_sizes; (void)n_in; (void)out_size; (void)ws_size;
    const float* out_t  = (const float*)d_in[0];
    const float* target = (const float*)d_in[1];
    const float* anch   = (const float*)d_in[2];

    char* w = (char*)d_ws;
    int*    tslot    = (int*)w;    w += (size_t)CAD * sizeof(int);
    int*    clsmask  = (int*)w;    w += (size_t)CAD * sizeof(int);
    float*  rec      = (float*)w;  w += (size_t)NB * MAXT * RECS * sizeof(float);
    float*  tprep    = (float*)w;  w += (size_t)NB * MAXT * TPS  * sizeof(float);
    int*    nvalid   = (int*)w;    w += 256;
    double* partials = (double*)w;

    k_init   <<<(CAD + BLK - 1) / BLK, BLK, 0, stream>>>(tslot, clsmask);
    k_targets<<<1, 64, 0, stream>>>(out_t, target, anch, tslot, clsmask, rec, tprep, nvalid);
    k_main   <<<NBLK, BLK, 0, stream>>>(out_t, anch, tslot, clsmask, rec, tprep, nvalid, partials);
    k_final  <<<1, BLK, 0, stream>>>(partials, (float*)d_out);
}